// PlackettLuceLoss_3427383902891
// MI455X (gfx1250) — compile-verified
//
#include <hip/hip_runtime.h>
#include <stdint.h>

// Plackett-Luce loss, MI455X (gfx1250, wave32).
// B = 2,097,152 rows x H = 18. Memory-bound streaming reduction:
// ~340 MB traffic -> ~15 us floor @ 23.3 TB/s. No matmul structure -> no WMMA;
// bulk data movement uses the CDNA5 async-to-LDS path
// (global_load_async_to_lds_b128 th:TH_LOAD_NT + s_wait_asynccnt).

#define H_DIM 18
#define ROWS_PER_BLOCK 256
#define THREADS 256

// ---- CDNA5 async global->LDS copy (GVS mode: saddr base + 32-bit vgpr offset)
// Non-temporal hint: data is streamed exactly once (340 MB through 192 MB L2),
// so keep it from evicting reusable lines.
__device__ __forceinline__ void async_copy16(uint32_t lds_addr, uint32_t goff,
                                             const void* gbase) {
  asm volatile("global_load_async_to_lds_b128 %0, %1, %2 th:TH_LOAD_NT"
               :: "v"(lds_addr), "v"(goff), "s"(gbase)
               : "memory");
}

__device__ __forceinline__ void wait_async0() {
  asm volatile("s_wait_asynccnt 0" ::: "memory");
}

__global__ __launch_bounds__(THREADS)
void pl_rows_kernel(const float* __restrict__ scores,
                    const int* __restrict__ rankings,
                    const unsigned char* __restrict__ mask,  // jax bool = 1 byte
                    double* __restrict__ partial,
                    int* __restrict__ counts) {
  __shared__ float         sS[ROWS_PER_BLOCK * H_DIM];   // 18 KB
  __shared__ int           sR[ROWS_PER_BLOCK * H_DIM];   // 18 KB
  __shared__ unsigned char sM[ROWS_PER_BLOCK * H_DIM];   // 4.5 KB
  __shared__ double        redD[THREADS];
  __shared__ int           redI[THREADS];

  const int tid = threadIdx.x;
  const int blk = blockIdx.x;

  // Generic pointers to LDS carry the byte offset in the low 32 bits
  // (SHARED aperture occupies ADDR[63:32] on gfx1250).
  const uint32_t offS = (uint32_t)(uintptr_t)(&sS[0]);
  const uint32_t offR = (uint32_t)(uintptr_t)(&sR[0]);
  const uint32_t offM = (uint32_t)(uintptr_t)(&sM[0]);

  const char* gS = (const char*)scores   + (size_t)blk * (ROWS_PER_BLOCK * H_DIM * 4);
  const char* gR = (const char*)rankings + (size_t)blk * (ROWS_PER_BLOCK * H_DIM * 4);
  const char* gM = (const char*)mask     + (size_t)blk * (ROWS_PER_BLOCK * H_DIM);

  // Block slab is contiguous & 16B aligned: 256*18*4 = 18432 B = 1152 x b128
  // (= 4*256 + 128 chunks), mask slab 4608 B = 288 x b128 (= 256 + 32).
  // Fully static schedule: no exec-mask loop overhead, no intra-wave divergence
  // (the tails split on wave boundaries: tid<128 = waves 0-3, tid<32 = wave 0).
#pragma unroll
  for (int k = 0; k < 4; ++k) {
    const uint32_t i = (uint32_t)(tid + k * THREADS) * 16u;
    async_copy16(offS + i, i, gS);
    async_copy16(offR + i, i, gR);
  }
  if (tid < 128) {
    const uint32_t i = (uint32_t)(tid + 4 * THREADS) * 16u;
    async_copy16(offS + i, i, gS);
    async_copy16(offR + i, i, gR);
  }
  {
    const uint32_t i = (uint32_t)tid * 16u;
    async_copy16(offM + i, i, gM);
  }
  if (tid < 32) {
    const uint32_t i = (uint32_t)(tid + THREADS) * 16u;
    async_copy16(offM + i, i, gM);
  }
  wait_async0();     // this wave's async LDS writes complete
  __syncthreads();   // all waves' writes visible to everyone

  // ---- per-thread row processing (row stride 18 dwords: conflict-free banks,
  // since 18*t mod 64 hits 32 distinct banks across a wave32)
  float s[H_DIM];
  int   key[H_DIM];
  int   L = 0;
#pragma unroll
  for (int j = 0; j < H_DIM; ++j) {
    s[j] = sS[tid * H_DIM + j];
    const int rk = sR[tid * H_DIM + j];
    const int v  = (sM[tid * H_DIM + j] != 0) & (rk > 0);
    // Composite key reproduces stable argsort tie-break (rank, then index).
    key[j] = v ? ((rk << 5) | j) : 0x7FFFFFFF;
    L += v;
  }

  // ---- Batcher merge-exchange sorting network, n = 18 (t = 5), 82 CEs,
  // fully static indices -> arrays stay in VGPRs.
#pragma unroll
  for (int pp = 0; pp < 5; ++pp) {
    const int p = 16 >> pp;
#pragma unroll
    for (int m = 0; m <= pp; ++m) {
      const int d = (m == 0) ? p : (32 >> m) - p;
      const int r = (m == 0) ? 0 : p;
#pragma unroll
      for (int i = 0; i < H_DIM - d; ++i) {
        if ((i & p) == r) {
          const int   ka = key[i], kb = key[i + d];
          const bool  sw = ka > kb;
          key[i] = sw ? kb : ka;  key[i + d] = sw ? ka : kb;
          const float sa = s[i], sb = s[i + d];
          s[i] = sw ? sb : sa;    s[i + d] = sw ? sa : sb;
        }
      }
    }
  }

  // ---- suffix logsumexp over ranked prefix; NEG padding == exp()->0, matches ref.
  float M = -3.402823466e38f;
#pragma unroll
  for (int i = 0; i < H_DIM; ++i) M = (i < L) ? fmaxf(M, s[i]) : M;

  float T = 0.0f, acc = 0.0f;
#pragma unroll
  for (int i = H_DIM - 1; i >= 0; --i) {
    T   += (i < L)     ? __expf(s[i] - M) : 0.0f;              // suffix sum incl. i
    acc += (i < L - 1) ? ((M + __logf(T)) - s[i]) : 0.0f;      // -(s_i - rev_lse_i)
  }

  const bool inc = (L >= 2);
  redD[tid] = inc ? (double)acc : 0.0;
  redI[tid] = inc ? 1 : 0;
  __syncthreads();
#pragma unroll
  for (int o = THREADS / 2; o > 0; o >>= 1) {
    if (tid < o) { redD[tid] += redD[tid + o]; redI[tid] += redI[tid + o]; }
    __syncthreads();
  }
  if (tid == 0) { partial[blk] = redD[0]; counts[blk] = redI[0]; }
}

__global__ __launch_bounds__(THREADS)
void pl_reduce_kernel(const double* __restrict__ partial,
                      const int* __restrict__ counts,
                      int nblocks, float* __restrict__ out) {
  __shared__ double sd[THREADS];
  __shared__ int    si[THREADS];
  double t = 0.0; int c = 0;
  for (int i = threadIdx.x; i < nblocks; i += THREADS) { t += partial[i]; c += counts[i]; }
  sd[threadIdx.x] = t; si[threadIdx.x] = c;
  __syncthreads();
#pragma unroll
  for (int o = THREADS / 2; o > 0; o >>= 1) {
    if (threadIdx.x < o) { sd[threadIdx.x] += sd[threadIdx.x + o]; si[threadIdx.x] += si[threadIdx.x + o]; }
    __syncthreads();
  }
  if (threadIdx.x == 0)
    out[0] = (si[0] > 0) ? (float)(sd[0] / (double)si[0]) : 0.0f;
}

extern "C" void kernel_launch(void* const* d_in, const int* in_sizes, int n_in,
                              void* d_out, int out_size, void* d_ws, size_t ws_size,
                              hipStream_t stream) {
  (void)n_in; (void)out_size; (void)ws_size;
  const float*         scores   = (const float*)d_in[0];
  const int*           rankings = (const int*)d_in[1];
  const unsigned char* mask     = (const unsigned char*)d_in[2];  // jax bool

  const int nRows   = in_sizes[0] / H_DIM;       // 2,097,152
  const int nBlocks = nRows / ROWS_PER_BLOCK;    // 8,192 (B divisible by 256)

  double* partial = (double*)d_ws;
  int*    counts  = (int*)((char*)d_ws + (size_t)nBlocks * sizeof(double));

  pl_rows_kernel<<<nBlocks, THREADS, 0, stream>>>(scores, rankings, mask, partial, counts);
  pl_reduce_kernel<<<1, THREADS, 0, stream>>>(partial, counts, nBlocks, (float*)d_out);
}